// MultiModalSDTPS_38637525794876
// MI455X (gfx1250) — compile-verified
//
#include <hip/hip_runtime.h>
#include <math.h>

#define NB   256   // batch
#define NTOK 128   // tokens per row
#define NC   512   // channels
#define KEEP 64    // tokens kept

#define USE_ASYNC_LDS 1

typedef __attribute__((ext_vector_type(2))) float v2f;
typedef __attribute__((ext_vector_type(8))) float v8f;
typedef __attribute__((ext_vector_type(2))) int   v2i;

#if USE_ASYNC_LDS && __has_builtin(__builtin_amdgcn_global_load_async_to_lds_b64)
#define HAVE_ASYNC 1
typedef __attribute__((address_space(1))) v2i as1_v2i;   // global-side param
typedef __attribute__((address_space(3))) v2i as3_v2i;   // LDS-side param
#else
#define HAVE_ASYNC 0
#endif

__device__ __forceinline__ void wait_async0() {
#if __has_builtin(__builtin_amdgcn_s_wait_asynccnt)
    __builtin_amdgcn_s_wait_asynccnt(0);
#else
    asm volatile("s_wait_asynccnt 0" ::: "memory");
#endif
}

__global__ __launch_bounds__(256) void sdtps_fused_kernel(
    const float* __restrict__ rgb_t, const float* __restrict__ nir_t,
    const float* __restrict__ tir_t, const float* __restrict__ rgb_g,
    const float* __restrict__ nir_g, const float* __restrict__ tir_g,
    float* __restrict__ out)
{
    const int bx = blockIdx.x;
    const int m  = bx >> 8;     // modality 0..2
    const int b  = bx & 255;    // batch index

    // ternary selects keep addrspace(1) inference -> global_load_*, not flat_load_*
    const float* T = (m == 0) ? rgb_t : ((m == 1) ? nir_t : tir_t);
    T += (size_t)b * NTOK * NC;

    const int tid  = threadIdx.x;
    const int w    = tid >> 5;   // wave id 0..7
    const int lane = tid & 31;

    __shared__ float sG[4][NC];        // raw globals; row 3 = zeros (WMMA B pad)
    __shared__ float sGinv[3];         // 1/max(||g||,eps)
    __shared__ float sDot[NTOK][4];    // raw dots vs 3 globals (padded)
    __shared__ float sInvT[NTOK];      // 1/max(||t||,eps)
    __shared__ float sS[3][NTOK];      // cosine scores
    __shared__ float sMn[3], sMx[3];
    __shared__ float sKey[NTOK];
    __shared__ int   sIdx[NTOK];
    __shared__ float sW[NTOK];
    __shared__ float sSum;

    // ---- stage the three global vectors into LDS (1 float2 per thread per row) ----
    {
        const size_t goff = (size_t)b * NC;
        const float2* g0 = (const float2*)(rgb_g + goff);
        const float2* g1 = (const float2*)(nir_g + goff);
        const float2* g2 = (const float2*)(tir_g + goff);
        float2* l0 = (float2*)&sG[0][0];
        float2* l1 = (float2*)&sG[1][0];
        float2* l2 = (float2*)&sG[2][0];
#if HAVE_ASYNC
        __builtin_amdgcn_global_load_async_to_lds_b64((as1_v2i*)&g0[tid], (as3_v2i*)&l0[tid], 0, 0);
        __builtin_amdgcn_global_load_async_to_lds_b64((as1_v2i*)&g1[tid], (as3_v2i*)&l1[tid], 0, 0);
        __builtin_amdgcn_global_load_async_to_lds_b64((as1_v2i*)&g2[tid], (as3_v2i*)&l2[tid], 0, 0);
        wait_async0();
#else
        l0[tid] = g0[tid];
        l1[tid] = g1[tid];
        l2[tid] = g2[tid];
#endif
        ((float2*)&sG[3][0])[tid] = make_float2(0.f, 0.f);   // zero pad row
    }
    __syncthreads();

    // ---- global-vector inverse norms: waves 0..2, wave32 shuffle reduce ----
    if (w < 3) {
        float ss = 0.f;
        for (int c = lane; c < NC; c += 32) { float v = sG[w][c]; ss += v * v; }
        for (int off = 16; off > 0; off >>= 1) ss += __shfl_xor(ss, off, 32);
        if (lane == 0) sGinv[w] = 1.0f / fmaxf(sqrtf(ss), 1e-12f);
    }
    __syncthreads();

    // ---- score GEMM: each wave owns a 16-token M-tile; K-loop over 512 ----
    // WMMA f32 16x16x4 A layout: lane = (kg<<4)|r holds M=r, K=kg*2+{0,1}
    const int R  = w * 16;
    const int r  = lane & 15;
    const int kg = lane >> 4;
    const int gj = (r < 3) ? r : 3;                 // B col -> sG row (3 = zeros)
    const float* rowp = T + (size_t)(R + r) * NC + kg * 2;
    const float* gcol = &sG[gj][kg * 2];
    float ssq = 0.f;

#if __has_builtin(__builtin_amdgcn_wmma_f32_16x16x4_f32)
    v8f acc0 = {}, acc1 = {};                       // two chains -> pipelined WMMAs
    for (int kc = 0; kc < NC; kc += 8) {
        v2f a0, a1, b0, b1;
        a0.x = rowp[kc];     a0.y = rowp[kc + 1];
        a1.x = rowp[kc + 4]; a1.y = rowp[kc + 5];
        b0.x = gcol[kc];     b0.y = gcol[kc + 1];
        b1.x = gcol[kc + 4]; b1.y = gcol[kc + 5];
        ssq += a0.x * a0.x + a0.y * a0.y + a1.x * a1.x + a1.y * a1.y;
        acc0 = __builtin_amdgcn_wmma_f32_16x16x4_f32(
            false, a0, false, b0, (short)0, acc0, false, false);
        acc1 = __builtin_amdgcn_wmma_f32_16x16x4_f32(
            false, a1, false, b1, (short)0, acc1, false, false);
    }
    // C layout: VGPR i -> lanes 0-15: (M=i, N=lane); lanes 16-31: (M=8+i, N=lane-16)
    #pragma unroll
    for (int i = 0; i < 8; i++) {
        int row = (lane < 16) ? i : (8 + i);
        if (r < 3) sDot[R + row][r] = acc0[i] + acc1[i];
    }
#else
    // VALU fallback (same memory pattern, lane-pair split over K mod 4)
    float d0 = 0.f, d1 = 0.f, d2 = 0.f;
    for (int kc = 0; kc < NC; kc += 4) {
        float ax = rowp[kc], ay = rowp[kc + 1];
        ssq += ax * ax + ay * ay;
        int k0 = kc + kg * 2;
        d0 += ax * sG[0][k0] + ay * sG[0][k0 + 1];
        d1 += ax * sG[1][k0] + ay * sG[1][k0 + 1];
        d2 += ax * sG[2][k0] + ay * sG[2][k0 + 1];
    }
    d0 += __shfl_down(d0, 16, 32);
    d1 += __shfl_down(d1, 16, 32);
    d2 += __shfl_down(d2, 16, 32);
    if (lane < 16) { sDot[R + r][0] = d0; sDot[R + r][1] = d1; sDot[R + r][2] = d2; }
#endif
    ssq += __shfl_down(ssq, 16, 32);                // combine the two K-half partials
    if (lane < 16) sInvT[R + r] = 1.0f / fmaxf(sqrtf(ssq), 1e-12f);
    __syncthreads();

    // ---- cosine scores ----
    if (tid < NTOK) {
        float it = sInvT[tid];
        #pragma unroll
        for (int j = 0; j < 3; j++)
            sS[j][tid] = sDot[tid][j] * it * sGinv[j];
    }
    __syncthreads();

    // ---- per-score min/max over the 128 tokens: waves 0..2 ----
    if (w < 3) {
        float mn = 3.4e38f, mx = -3.4e38f;
        for (int c = lane; c < NTOK; c += 32) {
            float v = sS[w][c];
            mn = fminf(mn, v);
            mx = fmaxf(mx, v);
        }
        for (int off = 16; off > 0; off >>= 1) {
            mn = fminf(mn, __shfl_xor(mn, off, 32));
            mx = fmaxf(mx, __shfl_xor(mx, off, 32));
        }
        if (lane == 0) { sMn[w] = mn; sMx[w] = mx; }
    }
    __syncthreads();

    // ---- fused score + sort keys ----
    if (tid < NTOK) {
        float sc = 0.f;
        #pragma unroll
        for (int j = 0; j < 3; j++)
            sc += (sS[j][tid] - sMn[j]) / (sMx[j] - sMn[j] + 1e-8f);
        sKey[tid] = sc * (1.0f / 3.0f);
        sIdx[tid] = tid;
    }
    __syncthreads();

    // ---- bitonic sort of 128 (key,idx) pairs, descending ----
    for (int k = 2; k <= NTOK; k <<= 1) {
        for (int j = k >> 1; j > 0; j >>= 1) {
            if (tid < NTOK) {
                int ixj = tid ^ j;
                if (ixj > tid) {
                    float a = sKey[tid], c = sKey[ixj];
                    bool dir = ((tid & k) == 0);
                    if ((a < c) == dir) {        // reversed comparator -> descending
                        sKey[tid] = c; sKey[ixj] = a;
                        int ia = sIdx[tid]; sIdx[tid] = sIdx[ixj]; sIdx[ixj] = ia;
                    }
                }
            }
            __syncthreads();
        }
    }

    // ---- softmax over sorted tail [KEEP,128): max is element KEEP ----
    if (tid >= KEEP && tid < NTOK)
        sW[tid] = expf(sKey[tid] - sKey[KEEP]);
    __syncthreads();
    if (w == 0) {
        float s = sW[KEEP + lane] + sW[KEEP + 32 + lane];
        for (int off = 16; off > 0; off >>= 1) s += __shfl_xor(s, off, 32);
        if (lane == 0) sSum = s;
    }
    __syncthreads();

    // ---- outputs: (3, 256, 65, 512) ----
    float* outBase = out + (size_t)(m * NB + b) * (KEEP + 1) * NC;

    // kept rows: one wave per row, float4 coalesced copy (hot in L2)
    for (int rr = w; rr < KEEP; rr += 8) {
        int src = sIdx[rr];
        const float4* sp = (const float4*)(T + (size_t)src * NC);
        float4* dp = (float4*)(outBase + (size_t)rr * NC);
        for (int c4 = lane; c4 < NC / 4; c4 += 32) dp[c4] = sp[c4];
    }

    // extra row: threads parallel over channels (float2 each -> coalesced per src row)
    float invSum = 1.0f / sSum;
    float2 acc2 = make_float2(0.f, 0.f);
    for (int p = KEEP; p < NTOK; p++) {
        float wgt = sW[p] * invSum;
        float2 v = ((const float2*)(T + (size_t)sIdx[p] * NC))[tid];
        acc2.x += wgt * v.x;
        acc2.y += wgt * v.y;
    }
    ((float2*)(outBase + (size_t)KEEP * NC))[tid] = acc2;
}

extern "C" void kernel_launch(void* const* d_in, const int* in_sizes, int n_in,
                              void* d_out, int out_size, void* d_ws, size_t ws_size,
                              hipStream_t stream) {
    (void)in_sizes; (void)n_in; (void)out_size; (void)d_ws; (void)ws_size;
    const float* rgb_t = (const float*)d_in[0];
    const float* nir_t = (const float*)d_in[1];
    const float* tir_t = (const float*)d_in[2];
    const float* rgb_g = (const float*)d_in[3];
    const float* nir_g = (const float*)d_in[4];
    const float* tir_g = (const float*)d_in[5];
    float* out = (float*)d_out;

    sdtps_fused_kernel<<<dim3(3 * NB), dim3(256), 0, stream>>>(
        rgb_t, nir_t, tir_t, rgb_g, nir_g, tir_g, out);
}